// NeRFAttention_40965398069382
// MI455X (gfx1250) — compile-verified
//
#include <hip/hip_runtime.h>
#include <hip/hip_bf16.h>

typedef float v2f __attribute__((ext_vector_type(2)));
typedef float v8f __attribute__((ext_vector_type(8)));

#define NRAY  16
#define HDIM  32
#define EMB   64
#define NF    256
#define NB    2
#define HW    1024
#define NTOK  6144            // SEQ * HW
#define NTILE 384             // NTOK / 16
#define MCH   16              // k-chunk (m rows) staged in LDS per step in k_av

// ---------------------------------------------------------------- norm scale
__global__ void k_scale(const float* __restrict__ scale,
                        const float* __restrict__ sW,
                        const float* __restrict__ sb,
                        float* __restrict__ ns) {
    int b = threadIdx.x;
    if (b < NB) {
        float v = scale[b*3+0]*sW[0] + scale[b*3+1]*sW[1] + scale[b*3+2]*sW[2] + sb[0];
        ns[b] = v * v;
    }
}

// -------------------------------------------- fused MLP (16->32->64) + LN
// grid: (24, 2 /*0=Q,1=K*/, 2 /*batch*/), block 256; one row per thread.
__global__ __launch_bounds__(256) void k_mlp(
    const float* __restrict__ rays,
    const float* __restrict__ qW1, const float* __restrict__ qb1,
    const float* __restrict__ qW2, const float* __restrict__ qb2,
    const float* __restrict__ qg,  const float* __restrict__ qbe,
    const float* __restrict__ kW1, const float* __restrict__ kb1,
    const float* __restrict__ kW2, const float* __restrict__ kb2,
    const float* __restrict__ kg,  const float* __restrict__ kbe,
    float* __restrict__ Qbuf, float* __restrict__ Kbuf) {

    __shared__ float sW1[HDIM*NRAY];
    __shared__ float sB1[HDIM];
    __shared__ float sW2[EMB*HDIM];
    __shared__ float sB2[EMB];
    __shared__ float sG[EMB];
    __shared__ float sBe[EMB];

    const int isK = blockIdx.y;
    const float* W1 = isK ? kW1 : qW1;  const float* B1 = isK ? kb1 : qb1;
    const float* W2 = isK ? kW2 : qW2;  const float* B2 = isK ? kb2 : qb2;
    const float* G  = isK ? kg  : qg;   const float* Be = isK ? kbe : qbe;

    const int tid = threadIdx.x;
    for (int i = tid; i < HDIM*NRAY; i += 256) sW1[i] = W1[i];
    for (int i = tid; i < EMB*HDIM;  i += 256) sW2[i] = W2[i];
    if (tid < HDIM) sB1[tid] = B1[tid];
    if (tid < EMB) { sB2[tid] = B2[tid]; sG[tid] = G[tid]; sBe[tid] = Be[tid]; }
    __syncthreads();

    const int b   = blockIdx.z;
    const int row = blockIdx.x * 256 + tid;           // < NTOK exactly
    const float* p = rays + ((size_t)b*NTOK + row)*NRAY;

    float pos[NRAY];
    #pragma unroll
    for (int i = 0; i < NRAY; ++i) pos[i] = p[i];

    float h[HDIM];
    #pragma unroll
    for (int i = 0; i < HDIM; ++i) {
        float a = sB1[i];
        #pragma unroll
        for (int j = 0; j < NRAY; ++j) a = fmaf(pos[j], sW1[i*NRAY+j], a);
        h[i] = a > 0.f ? a : 0.f;
    }

    float o[EMB]; float mu = 0.f;
    #pragma unroll
    for (int i = 0; i < EMB; ++i) {
        float a = sB2[i];
        #pragma unroll
        for (int j = 0; j < HDIM; ++j) a = fmaf(h[j], sW2[i*HDIM+j], a);
        o[i] = a; mu += a;
    }
    mu *= (1.f/EMB);
    float var = 0.f;
    #pragma unroll
    for (int i = 0; i < EMB; ++i) { float d = o[i]-mu; var = fmaf(d, d, var); }
    var *= (1.f/EMB);
    const float inv = rsqrtf(var + 1e-5f);
    const float sc  = isK ? 1.f : 0.125f;             // Q / sqrt(64)

    float* out = (isK ? Kbuf : Qbuf) + ((size_t)b*NTOK + row)*EMB;
    #pragma unroll
    for (int i = 0; i < EMB; ++i)
        out[i] = (sG[i]*(o[i]-mu)*inv + sBe[i]) * sc;
}

// ---------------------- scores: S = (Q K^T + mask) * ns, via WMMA f32 16x16x4
// grid: (NTILE*NTILE/8, 2), block 256 (8 waves); one 16x16 tile per wave.
__global__ __launch_bounds__(256) void k_scores(
    const float* __restrict__ Q, const float* __restrict__ K,
    const float* __restrict__ ns, float* __restrict__ attnw) {

    const int b    = blockIdx.y;
    const int wave = threadIdx.x >> 5;
    const int lane = threadIdx.x & 31;
    const int tile = blockIdx.x * 8 + wave;           // < NTILE*NTILE
    const int tm = tile % NTILE, tn = tile / NTILE;
    const int n0 = tn * 16, m0 = tm * 16;
    const int mrow = lane & 15, kh = lane >> 4;

    const float* qrow = Q + ((size_t)b*NTOK + n0 + mrow)*EMB + kh*2;
    const float* krow = K + ((size_t)b*NTOK + m0 + mrow)*EMB + kh*2;

    v8f c = {};
    #pragma unroll
    for (int kk = 0; kk < EMB/4; ++kk) {
        v2f a  = *(const v2f*)(qrow + kk*4);          // A: Q[n0+mrow, kk*4 + kh*2 +{0,1}]
        v2f bb = *(const v2f*)(krow + kk*4);          // B: K^T[k, m0+mrow] = K[m0+mrow, k]
        c = __builtin_amdgcn_wmma_f32_16x16x4_f32(
                false, a, false, bb, (short)0, c, false, false);
    }

    const float nsv = ns[b];
    const int col = m0 + mrow;
    const int vj = col >> 10, pj = col & (HW-1);
    float* outb = attnw + (size_t)b*NTOK*NTOK;
    #pragma unroll
    for (int r = 0; r < 8; ++r) {
        const int row = n0 + r + kh*8;
        const bool blocked = ((row >> 10) == vj) && ((row & (HW-1)) != pj);
        const float v = (c[r] + (blocked ? -1e12f : 0.f)) * nsv;
        outb[(size_t)row*NTOK + col] = v;
    }
}

// ------------------- column-wise online softmax reduction (softmax axis = 1)
// Each thread owns 4 adjacent columns -> each row step is one coalesced b128.
// grid: (NTOK/1024, 2), block 256.
__global__ __launch_bounds__(256) void k_colreduce(
    const float* __restrict__ attnw,
    float* __restrict__ cmax, float* __restrict__ csum) {
    const int b    = blockIdx.y;
    const int col4 = blockIdx.x * 256 + threadIdx.x;  // < NTOK/4
    const float4* S = (const float4*)(attnw + (size_t)b*NTOK*NTOK) + col4;
    float m0 = -3.402823466e38f, m1 = m0, m2 = m0, m3 = m0;
    float s0 = 0.f, s1 = 0.f, s2 = 0.f, s3 = 0.f;
    for (int r = 0; r < NTOK; ++r) {
        const float4 x = S[(size_t)r * (NTOK/4)];
        float t;
        t = fmaxf(m0, x.x); s0 = s0*expf(m0-t) + expf(x.x-t); m0 = t;
        t = fmaxf(m1, x.y); s1 = s1*expf(m1-t) + expf(x.y-t); m1 = t;
        t = fmaxf(m2, x.z); s2 = s2*expf(m2-t) + expf(x.z-t); m2 = t;
        t = fmaxf(m3, x.w); s3 = s3*expf(m3-t) + expf(x.w-t); m3 = t;
    }
    float4* cm = (float4*)(cmax + b*NTOK);
    float4* cs = (float4*)(csum + b*NTOK);
    cm[col4] = make_float4(m0, m1, m2, m3);
    cs[col4] = make_float4(s0, s1, s2, s3);
}

// ------------------------------------------- in-place normalize attn_w (b128)
// grid: (NTOK*NTOK/1024, 2), block 256; 4 elements per thread.
__global__ __launch_bounds__(256) void k_norm(
    float* __restrict__ attnw,
    const float* __restrict__ cmax, const float* __restrict__ csum) {
    const int b = blockIdx.y;
    const size_t i4 = (size_t)blockIdx.x * 256 + threadIdx.x;   // < NTOK*NTOK/4
    float4* S = (float4*)(attnw + (size_t)b*NTOK*NTOK);
    const int c4 = (int)(i4 % (NTOK/4));
    const float4 mx = ((const float4*)(cmax + b*NTOK))[c4];
    const float4 sm = ((const float4*)(csum + b*NTOK))[c4];
    float4 x = S[i4];
    x.x = expf(x.x - mx.x) / sm.x;
    x.y = expf(x.y - mx.y) / sm.y;
    x.z = expf(x.z - mx.z) / sm.z;
    x.w = expf(x.w - mx.w) / sm.w;
    S[i4] = x;
}

// --------------- attn_out = attn_w @ feats, WMMA f32 + async LDS staging.
// Block = 8 waves = 2 row-strips x 4 feature-quads; the whole block shares a
// 16(m) x 256(f) feats chunk staged in LDS with global_load_async_to_lds_b128
// (ASYNCcnt), double-buffered against the WMMA consumption.
// grid: (NTILE/2, 2), block 256.
__global__ __launch_bounds__(256) void k_av(
    const float* __restrict__ attnw, const float* __restrict__ feats,
    float* __restrict__ out) {

    __shared__ __align__(16) float sf[2][MCH*NF];     // 2 x 16 KB

    const int b    = blockIdx.y;
    const int tid  = threadIdx.x;
    const int wave = tid >> 5;
    const int lane = tid & 31;
    const int strip = blockIdx.x * 2 + (wave >> 2);   // 16-row strip
    const int quad  = wave & 3;                       // 64 feature columns
    const int n0 = strip * 16;
    const int mrow = lane & 15, kh = lane >> 4;

    const float* A = attnw + (size_t)b*NTOK*NTOK + (size_t)(n0 + mrow)*NTOK + kh*2;
    const float* F = feats + (size_t)b*NTOK*NF;

    // Stage feats rows [c*MCH, c*MCH+16) -> sf[c&1] : 1024 float4, 4 per thread.
    auto stage = [&](int c) {
        const char* src = (const char*)(F + (size_t)c*MCH*NF);
        const unsigned dst = (unsigned)(uintptr_t)&sf[c & 1][0];  // LDS byte addr
        #pragma unroll
        for (int j = 0; j < 4; ++j) {
            const int f4 = tid + j*256;                           // 0..1023
            unsigned long long ga = (unsigned long long)(uintptr_t)src
                                  + (unsigned long long)f4 * 16ull;
            unsigned la = dst + (unsigned)f4 * 16u;
            asm volatile("global_load_async_to_lds_b128 %0, %1, off"
                         :: "v"(la), "v"(ga) : "memory");
        }
    };

    stage(0);
    v8f acc[4] = {};
    for (int c = 0; c < NTOK/MCH; ++c) {
        asm volatile("s_wait_asynccnt 0" ::: "memory"); // own async stores done
        __syncthreads();                                // everyone's chunk visible
        if (c + 1 < NTOK/MCH) stage(c + 1);             // prefetch next chunk

        const float* buf = &sf[c & 1][0];
        #pragma unroll
        for (int mk = 0; mk < MCH/4; ++mk) {
            v2f a = *(const v2f*)(A + (c*MCH + mk*4));  // attn_w[n0+mrow, m+kh*2+{0,1}]
            const float* Bb = buf + (mk*4 + kh*2)*NF + quad*64 + mrow;
            #pragma unroll
            for (int t = 0; t < 4; ++t) {
                v2f bb;
                bb.x = Bb[t*16];                        // feats[m,   f0+mrow] (LDS)
                bb.y = Bb[t*16 + NF];                   // feats[m+1, f0+mrow] (LDS)
                acc[t] = __builtin_amdgcn_wmma_f32_16x16x4_f32(
                            false, a, false, bb, (short)0, acc[t], false, false);
            }
        }
        __syncthreads();       // all waves done reading sf[c&1] before restage
    }

    float* O = out + (size_t)b*NTOK*NF;
    #pragma unroll
    for (int t = 0; t < 4; ++t) {
        const int f0 = quad*64 + t*16;
        #pragma unroll
        for (int r = 0; r < 8; ++r) {
            const int row = n0 + r + kh*8;
            O[(size_t)row*NF + f0 + mrow] = acc[t][r];
        }
    }
}

// ---------------------------------------------------------------- launcher
extern "C" void kernel_launch(void* const* d_in, const int* in_sizes, int n_in,
                              void* d_out, int out_size, void* d_ws, size_t ws_size,
                              hipStream_t stream) {
    const float* rays  = (const float*)d_in[0];
    const float* feats = (const float*)d_in[1];
    const float* scale = (const float*)d_in[2];
    const float* qW1 = (const float*)d_in[3];  const float* qb1 = (const float*)d_in[4];
    const float* qW2 = (const float*)d_in[5];  const float* qb2 = (const float*)d_in[6];
    const float* qg  = (const float*)d_in[7];  const float* qbe = (const float*)d_in[8];
    const float* kW1 = (const float*)d_in[9];  const float* kb1 = (const float*)d_in[10];
    const float* kW2 = (const float*)d_in[11]; const float* kb2 = (const float*)d_in[12];
    const float* kg  = (const float*)d_in[13]; const float* kbe = (const float*)d_in[14];
    const float* sW  = (const float*)d_in[15]; const float* sb  = (const float*)d_in[16];

    float* attn_out = (float*)d_out;                          // [2,6144,256]
    float* attn_w   = attn_out + (size_t)NB*NTOK*NF;          // [2,6144,6144]

    float* ws   = (float*)d_ws;
    float* Qb   = ws;                                         // 2*6144*64
    float* Kb   = Qb + (size_t)NB*NTOK*EMB;
    float* cmax = Kb + (size_t)NB*NTOK*EMB;
    float* csum = cmax + NB*NTOK;
    float* ns   = csum + NB*NTOK;

    k_scale<<<1, 32, 0, stream>>>(scale, sW, sb, ns);
    k_mlp<<<dim3(NTOK/256, 2, NB), 256, 0, stream>>>(
        rays, qW1, qb1, qW2, qb2, qg, qbe, kW1, kb1, kW2, kb2, kg, kbe, Qb, Kb);
    k_scores<<<dim3(NTILE*NTILE/8, NB), 256, 0, stream>>>(Qb, Kb, ns, attn_w);
    k_colreduce<<<dim3(NTOK/1024, NB), 256, 0, stream>>>(attn_w, cmax, csum);
    k_norm<<<dim3((unsigned)((size_t)NTOK*NTOK/1024), NB), 256, 0, stream>>>(attn_w, cmax, csum);
    k_av<<<dim3(NTILE/2, NB), 256, 0, stream>>>(attn_w, feats, attn_out);
}